// InstanceGNN_45698452029992
// MI455X (gfx1250) — compile-verified
//
#include <hip/hip_runtime.h>
#include <hip/hip_bf16.h>
#include <math.h>

// ---------------------------------------------------------------------------
// GAT-based instance GNN for MI455X (gfx1250), wave32.
// Dense GEMMs use V_WMMA_F32_16X16X4_F32 with LDS-staged A and B tiles so the
// hot loop is two ds_load_2addr_b32 + one wmma, branch-free.
// Sparse edge phase: decomposed attention scores, fused exp+numerator+den
// scatter with f32 atomics (4 channels/thread), normalization deferred.
// ---------------------------------------------------------------------------

typedef float v2f __attribute__((ext_vector_type(2)));
typedef float v8f __attribute__((ext_vector_type(8)));

#define HEADS 4
#define HDIM 16
#define HID 64
#define ND 7
#define ED 6
#define HD 4
#define MKP 136   // match-head K padded (134 -> 136)

// ---- monotone float <-> uint encoding for atomicMax on floats -------------
__device__ __forceinline__ unsigned enc_f32(float f) {
    unsigned u = __float_as_uint(f);
    return (u & 0x80000000u) ? ~u : (u | 0x80000000u);
}
__device__ __forceinline__ float dec_f32(unsigned u) {
    u = (u & 0x80000000u) ? (u & 0x7FFFFFFFu) : ~u;
    return __uint_as_float(u);
}

// ---------------------------------------------------------------------------
// Zero-pad ragged-K weights into workspace (trailing rows are contiguous):
//   g1p : [8  x 64] from g1_Wn [7 x 64]
//   w1p : [136x 32] from mh_W1 [134 x 32]
// ---------------------------------------------------------------------------
__global__ void pad_weights_kernel(const float* __restrict__ g1_Wn,
                                   const float* __restrict__ mh_W1,
                                   float* __restrict__ g1p,
                                   float* __restrict__ w1p) {
    int idx = blockIdx.x * blockDim.x + threadIdx.x;
    if (idx < 8 * HID)    g1p[idx] = (idx < ND * HID) ? g1_Wn[idx] : 0.0f;
    if (idx < MKP * 32)   w1p[idx] = (idx < 134 * 32) ? mh_W1[idx] : 0.0f;
}

// ---------------------------------------------------------------------------
// fp32 WMMA GEMM: C[nrows x ncols] = A[nrows x K] * Bp[KP x ncols]
// Bp is zero-padded to KP rows (KP = 8 or 64, power of two; kps = log2 KP).
// A and B tiles staged in LDS (coalesced); inner loop is branch-free:
//   ds_load_2addr (A pair) + ds_load_2addr (B pair) + v_wmma.
// grid.x = row tiles of 16, grid.y = col tiles of 16, block = 32 (one wave).
// Fragments per ISA: A 16x4: lane%16 = M, VGPR v + 2*(lane/16) = K.
//                    B 4x16:  lane%16 = N, same K split.
//                    C/D:     VGPR r -> M = r + 8*(lane/16), N = lane%16.
// ---------------------------------------------------------------------------
__global__ void wmma_gemm_f32_kernel(const float* __restrict__ A, int lda,
                                     const float* __restrict__ Bp, int ldb,
                                     float* __restrict__ C, int ldc,
                                     int nrows, int K, int kps) {
    __shared__ float As[16 * 64];               // KP <= 64
    __shared__ float Bs[64 * 16];
    const int lane = threadIdx.x;
    const int half = lane >> 4;
    const int l16  = lane & 15;
    const int rowBase = blockIdx.x * 16;
    const int colBase = blockIdx.y * 16;
    const int col = colBase + l16;
    const int KP  = 1 << kps;

    // stage A tile (guards only in this cold path)
    for (int i = lane; i < (16 << kps); i += 32) {
        int r = i >> kps;
        int k = i & (KP - 1);
        int row = rowBase + r;
        float v = 0.0f;
        if (row < nrows && k < K) v = A[(size_t)row * lda + k];
        As[i] = v;
    }
    // stage B tile (fully in-range: Bp has KP valid rows, grid.y covers cols)
    for (int i = lane; i < (KP << 4); i += 32) {
        int k = i >> 4;
        Bs[i] = Bp[(size_t)k * ldb + colBase + (i & 15)];
    }
    __syncthreads();

    v8f c = {};
    const int abase = (l16 << kps) + 2 * half;
    const int bbase = (2 * half) * 16 + l16;
    for (int k0 = 0; k0 < KP; k0 += 4) {
        v2f a = *(const v2f*)&As[abase + k0];
        v2f b = { Bs[bbase + k0 * 16], Bs[bbase + k0 * 16 + 16] };
        c = __builtin_amdgcn_wmma_f32_16x16x4_f32(false, a, false, b,
                                                  (short)0, c, false, false);
    }

    const int mBase = rowBase + 8 * half;
    if (rowBase + 16 <= nrows) {                // full tile: straight stores
        float* p = C + (size_t)mBase * ldc + col;
#pragma unroll
        for (int r = 0; r < 8; ++r) {
            *p = c[r];
            p += ldc;
        }
    } else {                                    // tail tile
#pragma unroll
        for (int r = 0; r < 8; ++r) {
            int m = mBase + r;
            if (m < nrows) C[(size_t)m * ldc + col] = c[r];
        }
    }
}

// ---------------------------------------------------------------------------
// Per-node attention score precompute: ssrc[n,h] = dot(Wh[n,h,:], a_src[h,:])
// ---------------------------------------------------------------------------
__global__ void node_scores_kernel(const float* __restrict__ Wh,
                                   const float* __restrict__ a_src,
                                   const float* __restrict__ a_dst,
                                   float* __restrict__ ss,
                                   float* __restrict__ sd, int N) {
    int idx = blockIdx.x * blockDim.x + threadIdx.x;
    if (idx >= N * HEADS) return;
    int n = idx >> 2, h = idx & 3;
    const float* w = Wh + (size_t)n * HID + h * HDIM;
    float s0 = 0.f, s1 = 0.f;
#pragma unroll
    for (int k = 0; k < HDIM; ++k) {
        float v = w[k];
        s0 += v * a_src[h * HDIM + k];
        s1 += v * a_dst[h * HDIM + k];
    }
    ss[idx] = s0;
    sd[idx] = s1;
}

// ---------------------------------------------------------------------------
// Edge scores (leaky-relu) + global max (encoded-uint atomicMax)
// ---------------------------------------------------------------------------
__global__ void edge_score_kernel(const int* __restrict__ ei,
                                  const float* __restrict__ ef,
                                  const float* __restrict__ ss,
                                  const float* __restrict__ sd,
                                  const float* __restrict__ We, // [ED, HEADS]
                                  float* __restrict__ sc_out,   // [E, HEADS]
                                  unsigned* __restrict__ gmax, int E) {
    __shared__ float red[256];
    int e = blockIdx.x * blockDim.x + threadIdx.x;
    float m = -3.0e38f;
    if (e < E) {
        int s = ei[e];
        int d = ei[E + e];
        float ew[HEADS] = {0.f, 0.f, 0.f, 0.f};
#pragma unroll
        for (int k = 0; k < ED; ++k) {
            float v = ef[(size_t)e * ED + k];
#pragma unroll
            for (int h = 0; h < HEADS; ++h) ew[h] += v * We[k * HEADS + h];
        }
#pragma unroll
        for (int h = 0; h < HEADS; ++h) {
            float sc = ss[s * HEADS + h] + sd[d * HEADS + h] + ew[h];
            sc = (sc > 0.f) ? sc : 0.2f * sc;   // leaky_relu(0.2)
            sc_out[(size_t)e * HEADS + h] = sc;
            m = fmaxf(m, sc);
        }
    }
    red[threadIdx.x] = m;
    __syncthreads();
    for (int s = 128; s > 0; s >>= 1) {
        if (threadIdx.x < s) red[threadIdx.x] = fmaxf(red[threadIdx.x], red[threadIdx.x + s]);
        __syncthreads();
    }
    if (threadIdx.x == 0) atomicMax(gmax, enc_f32(red[0]));
}

// ---------------------------------------------------------------------------
// Fused softmax numerator + denominator scatter (normalization deferred):
//   one thread per (edge, 4-channel group):
//     acc[d, c..c+3] += Wh[s, c..c+3] * exp(sc[e,h] - gmax)   (b128 load)
//     den[d,h]       += exp(sc[e,h] - gmax)                   (one lane/head)
// Valid because sum_e Wh*(ex/den) == (sum_e Wh*ex)/den for fixed (d,h).
// ---------------------------------------------------------------------------
__global__ void edge_agg_kernel(const int* __restrict__ ei,
                                const float* __restrict__ scb,
                                const unsigned* __restrict__ gmax,
                                const float* __restrict__ Wh,
                                float* __restrict__ acc,
                                float* __restrict__ den, int E) {
    unsigned idx = blockIdx.x * blockDim.x + threadIdx.x;
    unsigned total = (unsigned)E * 16;          // 16 quads of 4 channels
    if (idx >= total) return;
    int e = idx >> 4;
    int q = idx & 15;           // quad index: channels c = 4q..4q+3
    int c = q << 2;
    int h = q >> 2;             // head = c / 16
    int s = ei[e];
    int d = ei[E + e];
    float gm = dec_f32(*gmax);
    float ex = expf(scb[(size_t)e * HEADS + h] - gm);
    float4 w = *(const float4*)&Wh[(size_t)s * HID + c];
    float* a = &acc[(size_t)d * HID + c];
    atomicAdd(a + 0, w.x * ex);
    atomicAdd(a + 1, w.y * ex);
    atomicAdd(a + 2, w.z * ex);
    atomicAdd(a + 3, w.w * ex);
    if ((q & 3) == 0) atomicAdd(&den[d * HEADS + h], ex);
}

// ---------------------------------------------------------------------------
// h_out = elu(acc / (den + 1e-9) + b)
// ---------------------------------------------------------------------------
__global__ void finalize_kernel(const float* __restrict__ acc,
                                const float* __restrict__ den,
                                const float* __restrict__ b,
                                float* __restrict__ hout, int N) {
    int idx = blockIdx.x * blockDim.x + threadIdx.x;
    if (idx >= N * HID) return;
    int n = idx >> 6, c = idx & 63, h = c >> 4;
    float x = acc[idx] / (den[n * HEADS + h] + 1e-9f) + b[c];
    hout[idx] = (x > 0.f) ? x : (expf(x) - 1.0f);
}

// ---------------------------------------------------------------------------
// Match head: ml = relu(z @ W1 + b1) @ W2 + b2 with z = [h2[i0]|h2[i1]|mef]
// z tile staged coalesced into LDS (16 x 136, zero-padded); W1 pre-padded.
// One wave per 16 rows; two 16-col WMMA tiles; W2 reduction via shfl_xor
// (masks <= 8 stay within each 16-lane half of the wave32).
// ---------------------------------------------------------------------------
__global__ void match_head_kernel(const float* __restrict__ h2,
                                  const int* __restrict__ mp,
                                  const float* __restrict__ mef,
                                  const float* __restrict__ w1p, // [136, 32] padded
                                  const float* __restrict__ b1,  // [32]
                                  const float* __restrict__ W2,  // [32]
                                  const float* __restrict__ b2,  // [1]
                                  float* __restrict__ ml, int M) {
    __shared__ float zt[16 * MKP];
    const int lane = threadIdx.x;
    const int half = lane >> 4;
    const int l16  = lane & 15;
    const int rowBase = blockIdx.x * 16;

    // stage z tile: coalesced row sweeps of h2 plus mef tail + zero pad
    for (int r = 0; r < 16; ++r) {
        int m = rowBase + r;
        if (m >= M) m = M - 1;
        int i0 = mp[m];
        int i1 = mp[M + m];
        float* row = &zt[r * MKP];
        row[lane]      = h2[(size_t)i0 * HID + lane];
        row[32 + lane] = h2[(size_t)i0 * HID + 32 + lane];
        row[64 + lane] = h2[(size_t)i1 * HID + lane];
        row[96 + lane] = h2[(size_t)i1 * HID + 32 + lane];
        if (lane < 8)
            row[128 + lane] = (lane < ED) ? mef[(size_t)m * ED + lane] : 0.0f;
    }
    __syncthreads();

    v8f c0 = {};
    v8f c1 = {};
    const int abase = l16 * MKP + 2 * half;
    for (int k0 = 0; k0 < MKP; k0 += 4) {
        v2f a = *(const v2f*)&zt[abase + k0];
        int ka = k0 + 2 * half;
        v2f bp = { w1p[ka * 32 + l16],      w1p[(ka + 1) * 32 + l16] };
        v2f bq = { w1p[ka * 32 + 16 + l16], w1p[(ka + 1) * 32 + 16 + l16] };
        c0 = __builtin_amdgcn_wmma_f32_16x16x4_f32(false, a, false, bp,
                                                   (short)0, c0, false, false);
        c1 = __builtin_amdgcn_wmma_f32_16x16x4_f32(false, a, false, bq,
                                                   (short)0, c1, false, false);
    }
    const float w2a = W2[l16], w2b = W2[16 + l16];
    const float ba  = b1[l16], bb  = b1[16 + l16];
    const float bias2 = b2[0];
#pragma unroll
    for (int r = 0; r < 8; ++r) {
        float v0 = c0[r] + ba; v0 = v0 > 0.f ? v0 : 0.f;
        float v1 = c1[r] + bb; v1 = v1 > 0.f ? v1 : 0.f;
        float pr = v0 * w2a + v1 * w2b;
        for (int mask = 1; mask < 16; mask <<= 1)
            pr += __shfl_xor(pr, mask, 32);
        if (l16 == 0) {
            int m = rowBase + r + 8 * half;
            if (m < M) ml[m] = pr + bias2;
        }
    }
}

// ---------------------------------------------------------------------------
// hhp[j, c] = sum_k hf[j,k] * hh_W1[HID+k, c]    (tiny: Nh x 32)
// ---------------------------------------------------------------------------
__global__ void hhp_kernel(const float* __restrict__ hf,
                           const float* __restrict__ W1, // [HID+HD, 32]
                           float* __restrict__ hhp, int Nh) {
    int idx = blockIdx.x * blockDim.x + threadIdx.x;
    if (idx >= Nh * 32) return;
    int j = idx >> 5, c = idx & 31;
    float s = 0.f;
#pragma unroll
    for (int k = 0; k < HD; ++k)
        s += hf[j * HD + k] * W1[(HID + k) * 32 + c];
    hhp[idx] = s;
}

// ---------------------------------------------------------------------------
// hl[n, j] = relu(hn[n,:] + hhp[j,:] + b1) @ W2 + b2 ; block = Nh threads
// ---------------------------------------------------------------------------
__global__ void hole_kernel(const float* __restrict__ hn,
                            const float* __restrict__ hhp,
                            const float* __restrict__ b1,
                            const float* __restrict__ W2,
                            const float* __restrict__ b2,
                            float* __restrict__ hl, int N, int Nh) {
    __shared__ float hs[32];
    __shared__ float bs[32];
    __shared__ float w2s[32];
    int n = blockIdx.x;
    int j = threadIdx.x;
    if (j < 32) {
        hs[j]  = hn[(size_t)n * 32 + j];
        bs[j]  = b1[j];
        w2s[j] = W2[j];
    }
    __syncthreads();
    if (n >= N || j >= Nh) return;
    float acc = 0.f;
#pragma unroll
    for (int k = 0; k < 32; ++k) {
        float v = hs[k] + hhp[j * 32 + k] + bs[k];
        v = v > 0.f ? v : 0.f;
        acc += v * w2s[k];
    }
    hl[(size_t)n * Nh + j] = acc + b2[0];
}

// ---------------------------------------------------------------------------
// Host launcher
// ---------------------------------------------------------------------------
extern "C" void kernel_launch(void* const* d_in, const int* in_sizes, int n_in,
                              void* d_out, int out_size, void* d_ws, size_t ws_size,
                              hipStream_t stream) {
    const float* nf    = (const float*)d_in[0];
    const int*   ei    = (const int*)  d_in[1];
    const float* ef    = (const float*)d_in[2];
    const int*   mp    = (const int*)  d_in[3];
    const float* mef   = (const float*)d_in[4];
    const float* hf    = (const float*)d_in[5];
    const float* g1_Wn = (const float*)d_in[6];
    const float* g1_We = (const float*)d_in[7];
    const float* g1_as = (const float*)d_in[8];
    const float* g1_ad = (const float*)d_in[9];
    const float* g1_b  = (const float*)d_in[10];
    const float* g2_Wn = (const float*)d_in[11];
    const float* g2_We = (const float*)d_in[12];
    const float* g2_as = (const float*)d_in[13];
    const float* g2_ad = (const float*)d_in[14];
    const float* g2_b  = (const float*)d_in[15];
    const float* mh_W1 = (const float*)d_in[16];
    const float* mh_b1 = (const float*)d_in[17];
    const float* mh_W2 = (const float*)d_in[18];
    const float* mh_b2 = (const float*)d_in[19];
    const float* hh_W1 = (const float*)d_in[20];
    const float* hh_b1 = (const float*)d_in[21];
    const float* hh_W2 = (const float*)d_in[22];
    const float* hh_b2 = (const float*)d_in[23];

    const int N  = in_sizes[0] / ND;
    const int E  = in_sizes[1] / 2;
    const int M  = in_sizes[3] / 2;
    const int Nh = in_sizes[5] / HD;

    // workspace carve-out (floats)
    float* ws = (float*)d_ws;
    size_t o = 0;
    float* Wh  = ws + o; o += (size_t)N * HID;
    float* ss  = ws + o; o += (size_t)N * HEADS;
    float* sd  = ws + o; o += (size_t)N * HEADS;
    float* den = ws + o; o += (size_t)N * HEADS;
    float* acc = ws + o; o += (size_t)N * HID;
    float* scb = ws + o; o += (size_t)E * HEADS;
    float* h1  = ws + o; o += (size_t)N * HID;
    float* h2  = ws + o; o += (size_t)N * HID;
    float* hn  = ws + o; o += (size_t)N * 32;
    float* hhp = ws + o; o += (size_t)Nh * 32;
    float* g1p = ws + o; o += 8 * HID;          // padded g1_Wn  [8 x 64]
    float* w1p = ws + o; o += (size_t)MKP * 32; // padded mh_W1 [136 x 32]
    unsigned* gmax = (unsigned*)(ws + o); o += 1;

    pad_weights_kernel<<<(MKP * 32 + 255) / 256, 256, 0, stream>>>(g1_Wn, mh_W1,
                                                                   g1p, w1p);

    auto gat_layer = [&](const float* X, int IN, const float* Bp, int kps,
                         const float* We, const float* as, const float* ad,
                         const float* b, float* hout) {
        dim3 gGemm((N + 15) / 16, HID / 16);
        wmma_gemm_f32_kernel<<<gGemm, 32, 0, stream>>>(X, IN, Bp, HID, Wh, HID,
                                                       N, IN, kps);

        int tns = N * HEADS;
        node_scores_kernel<<<(tns + 255) / 256, 256, 0, stream>>>(Wh, as, ad, ss, sd, N);

        hipMemsetAsync(den, 0, (size_t)N * HEADS * sizeof(float), stream);
        hipMemsetAsync(acc, 0, (size_t)N * HID * sizeof(float), stream);
        hipMemsetAsync(gmax, 0, sizeof(unsigned), stream);

        edge_score_kernel<<<(E + 255) / 256, 256, 0, stream>>>(ei, ef, ss, sd, We,
                                                               scb, gmax, E);
        unsigned tagg = (unsigned)E * 16;
        edge_agg_kernel<<<(tagg + 255) / 256, 256, 0, stream>>>(ei, scb, gmax, Wh,
                                                                acc, den, E);
        int tfin = N * HID;
        finalize_kernel<<<(tfin + 255) / 256, 256, 0, stream>>>(acc, den, b, hout, N);
    };

    gat_layer(nf, ND, g1p, 3, g1_We, g1_as, g1_ad, g1_b, h1);    // K=7, KP=8
    gat_layer(h1, HID, g2_Wn, 6, g2_We, g2_as, g2_ad, g2_b, h2); // K=64

    float* ml = (float*)d_out;        // [M]
    float* hl = ml + M;               // [N * Nh]

    match_head_kernel<<<(M + 15) / 16, 32, 0, stream>>>(h2, mp, mef, w1p, mh_b1,
                                                        mh_W2, mh_b2, ml, M);

    dim3 gHn((N + 15) / 16, 2);
    wmma_gemm_f32_kernel<<<gHn, 32, 0, stream>>>(h2, HID, hh_W1, 32, hn, 32,
                                                 N, HID, 6);

    hhp_kernel<<<(Nh * 32 + 255) / 256, 256, 0, stream>>>(hf, hh_W1, hhp, Nh);

    hole_kernel<<<N, Nh, 0, stream>>>(hn, hhp, hh_b1, hh_W2, hh_b2, hl, N, Nh);
}